// FCDecoder_21715354649470
// MI455X (gfx1250) — compile-verified
//
#include <hip/hip_runtime.h>

typedef __attribute__((ext_vector_type(2))) float v2f;
typedef __attribute__((ext_vector_type(8))) float v8f;

#define B_    4
#define N_    100000
#define C_    128
#define RES_  128
#define PLANE_PX   (RES_ * RES_)                 // 16384 pixels per (b, plane) image
#define IMG_STRIDE ((size_t)C_ * PLANE_PX)       // 2^21 floats per batch within one plane tensor

// ---------------------------------------------------------------------------
// Pass 1: S[q][b][y][x] = sum_ch fc_w[ch] * plane_q[b][ch][y][x]
//
// One wave32 per 16-pixel tile. K loop (128 channels, 4 per step) runs through
// v_wmma_f32_16x16x4_f32 with A = fc_w chunk replicated across all 16 M rows,
// B = plane values for the 16 pixels. Every row of D is then the per-pixel
// channel dot; lanes 0..15 extract row M=0 from d[0].
//
// ISA layouts used (CDNA5 ISA):
//   A 16x4 f32 : lanes 0-15 -> v0=K0, v1=K1 ; lanes 16-31 -> v0=K2, v1=K3
//   B 4x16 f32 : k = 2*(lane>=16) + vgpr, n = lane&15
//   D 16x16 f32: lanes 0-15, vgpr0 -> M=0, N=lane
// ---------------------------------------------------------------------------
__global__ void __launch_bounds__(256) reduce_planes_wmma(
    const float* __restrict__ cxz, const float* __restrict__ cxy,
    const float* __restrict__ cyz, const float* __restrict__ fcw,
    float* __restrict__ S)
{
    __shared__ float wl[C_];
    for (int t = threadIdx.x; t < C_; t += blockDim.x) wl[t] = fcw[t];
    __syncthreads();

    const int wave = threadIdx.x >> 5;
    const int lane = threadIdx.x & 31;
    const int half = lane >> 4;          // K half selector (ISA A/B striping)
    const int col  = lane & 15;          // pixel within tile (N index)

    const int tile    = blockIdx.x * 8 + wave;   // 12288 tiles total, exact fit
    const int img     = tile >> 10;              // 0..11  == q*B + b
    const int q       = img >> 2;                // plane index 0..2   (B_ == 4)
    const int b       = img & 3;                 // batch index
    const int pixbase = (tile & 1023) << 4;      // 16-pixel tile base in image

    const float* base = (q == 0) ? cxz : (q == 1) ? cxy : cyz;
    const float* ip   = base + (size_t)b * IMG_STRIDE + (size_t)pixbase + col;

    v8f acc = {};
    #pragma unroll
    for (int ch0 = 0; ch0 < C_; ch0 += 4) {
        const int ch = ch0 + 2 * half;
        v2f A, Bv;
        A[0]  = wl[ch];
        A[1]  = wl[ch + 1];
        Bv[0] = ip[(size_t)ch * PLANE_PX];
        Bv[1] = ip[(size_t)(ch + 1) * PLANE_PX];
        acc = __builtin_amdgcn_wmma_f32_16x16x4_f32(
            /*neg_a=*/false, A, /*neg_b=*/false, Bv,
            /*c_mod=*/(short)0, acc, /*reuse_a=*/false, /*reuse_b=*/false);
    }

    if (lane < 16)
        S[(size_t)img * PLANE_PX + pixbase + col] = acc[0];
}

// ---------------------------------------------------------------------------
// Pass 2: one thread per point. Bilinear-sample the three reduced planes
// (768 KB table, L2-resident), add the p-tail dot and bias.
// Matches the reference's normalize_coordinate + align_corners=True +
// border padding exactly.
// ---------------------------------------------------------------------------
__global__ void __launch_bounds__(256) sample_fc(
    const float* __restrict__ p, const float* __restrict__ S,
    const float* __restrict__ fcw, const float* __restrict__ fcb,
    float* __restrict__ out)
{
    const int i = blockIdx.x * blockDim.x + threadIdx.x;
    if (i >= B_ * N_) return;
    const int b = i / N_;

    const float p0 = p[3 * i + 0];
    const float p1 = p[3 * i + 1];
    const float p2 = p[3 * i + 2];

    const float inv = 1.0f / (1.0f + 0.1f + 1e-5f);   // 1/(1+PAD+EPS)
    float acc = 0.0f;

    #pragma unroll
    for (int q = 0; q < 3; ++q) {
        // plane axis selection: c_xz:(0,2), c_xy:(0,1), c_yz:(1,2)
        const float pa = (q == 2) ? p1 : p0;
        const float pb = (q == 1) ? p1 : p2;

        float u = fminf(fmaxf(pa * inv + 0.5f, 0.0f), 1.0f - 1e-5f);
        float v = fminf(fmaxf(pb * inv + 0.5f, 0.0f), 1.0f - 1e-5f);

        const float x  = u * (float)(RES_ - 1);
        const float y  = v * (float)(RES_ - 1);
        const float xf = floorf(x);
        const float yf = floorf(y);
        const float wx = x - xf;
        const float wy = y - yf;

        const int x0 = min(max((int)xf, 0), RES_ - 1);
        const int x1 = min(max((int)xf + 1, 0), RES_ - 1);
        const int y0 = min(max((int)yf, 0), RES_ - 1);
        const int y1 = min(max((int)yf + 1, 0), RES_ - 1);

        const float* Sq = S + (size_t)(q * B_ + b) * PLANE_PX;
        const float s00 = Sq[(y0 << 7) + x0];
        const float s01 = Sq[(y0 << 7) + x1];
        const float s10 = Sq[(y1 << 7) + x0];
        const float s11 = Sq[(y1 << 7) + x1];

        acc += (s00 * (1.0f - wx) + s01 * wx) * (1.0f - wy)
             + (s10 * (1.0f - wx) + s11 * wx) * wy;
    }

    out[i] = acc + p0 * fcw[C_ + 0] + p1 * fcw[C_ + 1] + p2 * fcw[C_ + 2] + fcb[0];
}

// ---------------------------------------------------------------------------
extern "C" void kernel_launch(void* const* d_in, const int* in_sizes, int n_in,
                              void* d_out, int out_size, void* d_ws, size_t ws_size,
                              hipStream_t stream) {
    const float* p   = (const float*)d_in[0];   // [B,N,3]
    const float* cxz = (const float*)d_in[1];   // [B,C,RES,RES]
    const float* cxy = (const float*)d_in[2];
    const float* cyz = (const float*)d_in[3];
    const float* fcw = (const float*)d_in[4];   // [C+3, 1]
    const float* fcb = (const float*)d_in[5];   // [1]
    float* out = (float*)d_out;                 // [B,N]
    float* S   = (float*)d_ws;                  // 3*B*RES*RES floats = 768 KB

    // Pass 1: 12288 wave-tiles, 8 waves per 256-thread block -> 1536 blocks.
    reduce_planes_wmma<<<1536, 256, 0, stream>>>(cxz, cxy, cyz, fcw, S);

    // Pass 2: one thread per point.
    const int total = B_ * N_;
    sample_fc<<<(total + 255) / 256, 256, 0, stream>>>(p, S, fcw, fcb, out);
}